// continuous_seg_171798692345
// MI455X (gfx1250) — compile-verified
//
#include <hip/hip_runtime.h>
#include <math.h>

typedef __attribute__((ext_vector_type(2))) float v2f;
typedef __attribute__((ext_vector_type(8))) float v8f;

#define STRH 130   // padded LDS stride for 128-float rows (bank-conflict avoidance)
#define STRZ 520   // padded LDS stride for 512-float z rows

// ---------------------------------------------------------------------------
// Cell step: z = [x | h | h_after | h_before] @ W_cell + b ; LSTM gate math.
// One block per 16-row M tile (rows = flattened (b,s), tile never crosses b).
// 8 waves; each wave owns 4 N-tiles (4 live accumulators) so every A fragment
// from LDS feeds 4 independent v_wmma chains.
// ---------------------------------------------------------------------------
__global__ __launch_bounds__(256) void cell_kernel(
    const float* __restrict__ input, const float* __restrict__ WT,
    const float* __restrict__ bcell, const float* __restrict__ h_in,
    float* __restrict__ h_out, float* __restrict__ c, int t)
{
    __shared__ float sh_h[18 * STRH];   // rows s0-1 .. s0+16 of h (zero-padded)
    __shared__ float sh_x[16 * 8];      // 16 x 5 input slice
    __shared__ float sh_z[16 * STRZ];   // 16 x 512 gate pre-activations

    const int tid = threadIdx.x;
    const int R0  = blockIdx.x * 16;    // global row base (b*128 + s0)
    const int bb  = R0 >> 7;
    const int s0  = R0 & 127;

    // Stage h rows s0-1 .. s0+16 (zeros outside [0,128): matches reference pad)
    for (int i = tid; i < 18 * 128; i += 256) {
        int r = i >> 7, k = i & 127;
        int s = s0 - 1 + r;
        float v = 0.0f;
        if (s >= 0 && s < 128) v = h_in[(bb * 128 + s) * 128 + k];
        sh_h[r * STRH + k] = v;
    }
    // Stage x tile (16 rows x 5 features)
    if (tid < 16 * 5) {
        int m = tid / 5, k = tid % 5;
        sh_x[m * 8 + k] = input[(((bb * 128) + t) * 128 + (s0 + m)) * 5 + k];
    }
    __syncthreads();

    const int wave  = tid >> 5;
    const int lane  = tid & 31;
    const int lm    = lane & 15;       // N column within tile / A row
    const int hi    = lane >> 4;       // 0: K=0,1  1: K=2,3 (ISA A/B layout)
    const int kHalf = hi * 2;

    const float* wc0 = WT + ((wave * 4 + 0) * 16 + lm) * 392;
    const float* wc1 = WT + ((wave * 4 + 1) * 16 + lm) * 392;
    const float* wc2 = WT + ((wave * 4 + 2) * 16 + lm) * 392;
    const float* wc3 = WT + ((wave * 4 + 3) * 16 + lm) * 392;

    v8f acc0 = {}, acc1 = {}, acc2 = {}, acc3 = {};

    // x part (K = 0..4): plain FMAs, consistent with the WMMA C/D layout
    #pragma unroll
    for (int v = 0; v < 8; ++v) {
        int m = v + hi * 8;
        float x0 = sh_x[m*8+0], x1 = sh_x[m*8+1], x2 = sh_x[m*8+2],
              x3 = sh_x[m*8+3], x4 = sh_x[m*8+4];
        acc0[v] += x0*wc0[0] + x1*wc0[1] + x2*wc0[2] + x3*wc0[3] + x4*wc0[4];
        acc1[v] += x0*wc1[0] + x1*wc1[1] + x2*wc1[2] + x3*wc1[3] + x4*wc1[4];
        acc2[v] += x0*wc2[0] + x1*wc2[1] + x2*wc2[2] + x3*wc2[3] + x4*wc2[4];
        acc3[v] += x0*wc3[0] + x1*wc3[1] + x2*wc3[2] + x3*wc3[3] + x4*wc3[4];
    }

    // Recurrent parts: h (W rows 5..132), h_after (133..260), h_before (261..388)
    #pragma unroll
    for (int src = 0; src < 3; ++src) {
        const int shBase = (src == 0) ? 1 : (src == 1) ? 2 : 0;
        const float* arow = &sh_h[(shBase + lm) * STRH];
        const int wOff = 5 + src * 128;
        #pragma unroll 4
        for (int kk = 0; kk < 128; kk += 4) {
            v2f a;
            a.x = arow[kk + kHalf];
            a.y = arow[kk + kHalf + 1];
            v2f b0, b1, b2, b3;
            b0.x = wc0[wOff + kk + kHalf]; b0.y = wc0[wOff + kk + kHalf + 1];
            b1.x = wc1[wOff + kk + kHalf]; b1.y = wc1[wOff + kk + kHalf + 1];
            b2.x = wc2[wOff + kk + kHalf]; b2.y = wc2[wOff + kk + kHalf + 1];
            b3.x = wc3[wOff + kk + kHalf]; b3.y = wc3[wOff + kk + kHalf + 1];
            acc0 = __builtin_amdgcn_wmma_f32_16x16x4_f32(
                        false, a, false, b0, (short)0, acc0, false, false);
            acc1 = __builtin_amdgcn_wmma_f32_16x16x4_f32(
                        false, a, false, b1, (short)0, acc1, false, false);
            acc2 = __builtin_amdgcn_wmma_f32_16x16x4_f32(
                        false, a, false, b2, (short)0, acc2, false, false);
            acc3 = __builtin_amdgcn_wmma_f32_16x16x4_f32(
                        false, a, false, b3, (short)0, acc3, false, false);
        }
    }

    #pragma unroll
    for (int v = 0; v < 8; ++v) {
        int m = v + hi * 8;
        sh_z[m * STRZ + (wave * 4 + 0) * 16 + lm] = acc0[v];
        sh_z[m * STRZ + (wave * 4 + 1) * 16 + lm] = acc1[v];
        sh_z[m * STRZ + (wave * 4 + 2) * 16 + lm] = acc2[v];
        sh_z[m * STRZ + (wave * 4 + 3) * 16 + lm] = acc3[v];
    }
    __syncthreads();

    // Gate math: i,f,g,o splits of z; c in-place (rows are block-private)
    for (int i = tid; i < 2048; i += 256) {
        int m = i >> 7, j = i & 127;
        float zi = sh_z[m*STRZ + j]       + bcell[j];
        float zf = sh_z[m*STRZ + 128 + j] + bcell[128 + j];
        float zg = sh_z[m*STRZ + 256 + j] + bcell[256 + j];
        float zo = sh_z[m*STRZ + 384 + j] + bcell[384 + j];
        int row = R0 + m;
        float cp = c[row*128 + j];
        float si = 1.0f / (1.0f + __expf(-zi));
        float sf = 1.0f / (1.0f + __expf(-zf));
        float so = 1.0f / (1.0f + __expf(-zo));
        float cn = sf * cp + si * tanhf(zg);
        c[row*128 + j]     = cn;
        h_out[row*128 + j] = so * tanhf(cn);
    }
}

// ---------------------------------------------------------------------------
// Output head: hidden = relu(h @ W1 + b1) via WMMA; out = hidden @ W2 + b2
// (N=3 second GEMM done as 48 per-thread dot products from LDS).
// ---------------------------------------------------------------------------
__global__ __launch_bounds__(256) void out_kernel(
    const float* __restrict__ h, const float* __restrict__ W1T,
    const float* __restrict__ b1, const float* __restrict__ W2T,
    const float* __restrict__ b2, float* __restrict__ outbuf)
{
    __shared__ float sh_a[16 * STRH];
    __shared__ float sh_hid[16 * STRH];
    const int tid = threadIdx.x;
    const int R0  = blockIdx.x * 16;

    for (int i = tid; i < 2048; i += 256) {
        int m = i >> 7, k = i & 127;
        sh_a[m*STRH + k] = h[(R0 + m)*128 + k];
    }
    __syncthreads();

    const int wave = tid >> 5, lane = tid & 31;
    const int lm = lane & 15, hi = lane >> 4, kHalf = hi * 2;
    const int col = wave * 16 + lm;
    const float* wk   = W1T + col * 128;
    const float* arow = &sh_a[lm * STRH];

    v8f acc = {};
    #pragma unroll 8
    for (int kk = 0; kk < 128; kk += 4) {
        v2f a, b;
        a.x = arow[kk + kHalf]; a.y = arow[kk + kHalf + 1];
        b.x = wk[kk + kHalf];   b.y = wk[kk + kHalf + 1];
        acc = __builtin_amdgcn_wmma_f32_16x16x4_f32(
                false, a, false, b, (short)0, acc, false, false);
    }
    float bias = b1[col];
    #pragma unroll
    for (int v = 0; v < 8; ++v) {
        float x = acc[v] + bias;
        sh_hid[(v + hi * 8) * STRH + col] = x > 0.0f ? x : 0.0f;
    }
    __syncthreads();

    if (tid < 48) {
        int m = tid / 3, cc = tid % 3;
        const float* hr = &sh_hid[m * STRH];
        const float* w2 = W2T + cc * 128;
        float s = 0.0f;
        for (int k = 0; k < 128; ++k) s += hr[k] * w2[k];
        outbuf[(R0 + m) * 3 + cc] = s + b2[cc];
    }
}

// ---------------------------------------------------------------------------
// _next_input: mix out with xt/xn, write to final output at step t (t >= 4).
// ---------------------------------------------------------------------------
__global__ __launch_bounds__(256) void nd_kernel(
    const float* __restrict__ input, const float* __restrict__ outbuf,
    float* __restrict__ out, int t)
{
    int gid = blockIdx.x * 256 + threadIdx.x;
    if (gid >= 1024) return;
    int bb = gid >> 7, s = gid & 127;
    const float* o = &outbuf[gid * 3];
    float n0 = o[0];
    float n1, sp0, sp1;
    if (s == 0) {
        const float* xn0 = &input[((bb * 128 + (t + 1)) * 128 + 0) * 5];
        n1 = xn0[1]; sp0 = xn0[3]; sp1 = xn0[4];
    } else {
        n1  = outbuf[(gid - 1) * 3 + 0];
        sp0 = o[1]; sp1 = o[2];
    }
    float fn = input[((bb * 128 + t) * 128 + s) * 5 + 2];
    float n2 = fn + n1 - n0;
    float* dst = &out[((bb * 123 + (t - 4)) * 128 + s) * 5];
    dst[0] = n0; dst[1] = n1; dst[2] = n2; dst[3] = sp0; dst[4] = sp1;
}

// ---------------------------------------------------------------------------
// Prep: zero h0/c; transpose W_cell -> WT[512][392] (K-padded), W1 -> W1T,
// W2 -> W2T so B-fragment loads are contiguous b64.
// ---------------------------------------------------------------------------
__global__ __launch_bounds__(256) void zero_kernel(float* p, int n) {
    int i = blockIdx.x * 256 + threadIdx.x;
    if (i < n) p[i] = 0.0f;
}

__global__ __launch_bounds__(256) void prep_kernel(
    const float* __restrict__ Wc, const float* __restrict__ W1,
    const float* __restrict__ W2, float* __restrict__ WT,
    float* __restrict__ W1T, float* __restrict__ W2T)
{
    int i = blockIdx.x * 256 + threadIdx.x;
    if (i < 512 * 392) {
        int col = i / 392, k = i % 392;
        WT[i] = (k < 389) ? Wc[k * 512 + col] : 0.0f;
    } else if (i < 512 * 392 + 16384) {
        int j = i - 512 * 392;
        int col = j >> 7, k = j & 127;
        W1T[j] = W1[k * 128 + col];
    } else if (i < 512 * 392 + 16384 + 384) {
        int j = i - 512 * 392 - 16384;
        int cc = j >> 7, k = j & 127;
        W2T[j] = W2[k * 3 + cc];
    }
}

extern "C" void kernel_launch(void* const* d_in, const int* in_sizes, int n_in,
                              void* d_out, int out_size, void* d_ws, size_t ws_size,
                              hipStream_t stream)
{
    (void)in_sizes; (void)n_in; (void)out_size; (void)ws_size;
    const float* input = (const float*)d_in[0];
    const float* Wc    = (const float*)d_in[1];
    const float* bc    = (const float*)d_in[2];
    const float* W1    = (const float*)d_in[3];
    const float* b1    = (const float*)d_in[4];
    const float* W2    = (const float*)d_in[5];
    const float* b2    = (const float*)d_in[6];
    float* ws   = (float*)d_ws;
    float* WT   = ws;                    // 512*392       = 200704
    float* W1T  = ws + 200704;           // 128*128       =  16384
    float* W2T  = ws + 217088;           // 3*128 (pad)   =    384
    float* hbuf = ws + 217472;           // 2 x 1024*128  = 262144
    float* cbuf = ws + 479616;           // 1024*128      = 131072
    float* obuf = ws + 610688;           // 1024*3        =   3072
    float* out  = (float*)d_out;

    // zero h(ping+pong) and c in one contiguous pass (393216 floats)
    zero_kernel<<<393216 / 256, 256, 0, stream>>>(hbuf, 393216);
    prep_kernel<<<(217472 + 255) / 256, 256, 0, stream>>>(Wc, W1, W2, WT, W1T, W2T);

    for (int t = 0; t < 127; ++t) {
        const float* h_in = hbuf + (t & 1) * 131072;
        float* h_out      = hbuf + ((t + 1) & 1) * 131072;
        cell_kernel<<<64, 256, 0, stream>>>(input, WT, bc, h_in, h_out, cbuf, t);
        out_kernel<<<64, 256, 0, stream>>>(h_out, W1T, b1, W2T, b2, obuf);
        if (t >= 4) nd_kernel<<<4, 256, 0, stream>>>(input, obuf, out, t);
    }
}